// SymmetricFeatures_23158463660452
// MI455X (gfx1250) — compile-verified
//
#include <hip/hip_runtime.h>
#include <cstdint>

typedef float v2f __attribute__((ext_vector_type(2)));
typedef float v8f __attribute__((ext_vector_type(8)));

#define WMMA_F32X4(A, Bv, C) \
    __builtin_amdgcn_wmma_f32_16x16x4_f32(false, (A), false, (Bv), (short)0, (C), false, false)

// Branchless tanh: tanh(x) = 1 - 2/(exp(2x)+1).
// v_exp_f32 + v_rcp_f32, no EXEC divergence, saturates correctly at +/-inf.
__device__ __forceinline__ float fast_tanh(float x) {
    float e = __expf(2.0f * x);
    return 1.0f - 2.0f * __builtin_amdgcn_rcpf(e + 1.0f);
}

namespace {
constexpr int NB   = 1024;  // batch
constexpr int NEL  = 32;    // electrons
constexpr int NION = 8;
constexpr int D1   = 256;   // d_one
constexpr int D2   = 32;    // d_two
constexpr int DION = 64;    // d_ion
constexpr int DEI  = 32;    // d_elion
constexpr int EMB  = 64;
constexpr int F    = 928;   // feature width: 256*3 + 32*2 + 64 + 32
}

__global__ __launch_bounds__(256) void symfeat_kernel(
    const float* __restrict__ h_one,
    const float* __restrict__ h_ion,
    const float* __restrict__ h_el_el,
    const float* __restrict__ h_el_ion,
    const float* __restrict__ W_hmap,
    const float* __restrict__ b_hmap,
    const float* __restrict__ W_wsame,
    const float* __restrict__ b_wsame,
    const float* __restrict__ W_wdiff,
    const float* __restrict__ b_wdiff,
    const float* __restrict__ W_ion,
    const float* __restrict__ b_ion,
    float* __restrict__ feats,
    float* __restrict__ mult_out)
{
    __shared__ float sWsame[D2 * EMB];
    __shared__ float sWdiff[D2 * EMB];
    __shared__ float sBias[2][EMB];
    __shared__ float sHmap[NEL * EMB];   // tanh(h_one @ W_hmap + b)
    __shared__ float sHion[NION * DEI];  // tanh(h_ion @ W_ion + b)
    __shared__ float sAcc[NEL * EMB];    // sum_j el_el_mult[i,j,n]

    const int b    = blockIdx.x;
    const int tid  = threadIdx.x;
    const int lane = tid & 31;
    const int wave = tid >> 5;
    const int l15  = lane & 15;
    const int half = lane >> 4;          // 0: lanes 0-15, 1: lanes 16-31

    // ---------- phase 0: stage pair weights, zero el_el accumulator ----------
    for (int idx = tid; idx < D2 * EMB; idx += 256) {
        sWsame[idx] = W_wsame[idx];
        sWdiff[idx] = W_wdiff[idx];
        sAcc[idx]   = 0.0f;              // NEL*EMB == D2*EMB == 2048
    }
    if (tid < EMB) {
        sBias[0][tid] = b_wsame[tid];
        sBias[1][tid] = b_wdiff[tid];
    }
    __syncthreads();

    const float* h1 = h_one + (size_t)b * NEL * D1;
    float*       fb = feats + (size_t)b * NEL * F;

    // ---------- phase 1a: stream h_one: copy + spin-averages + tiled sections ----------
    {
        const int c = tid;               // 0..255 == D1
        float au = 0.0f, ad = 0.0f;
        for (int e = 0; e < NEL; ++e) {
            float v = h1[e * D1 + c];
            fb[e * F + c] = v;           // feats[:,0:256] = h_one
            if (e < 16) au += v; else ad += v;
        }
        au *= (1.0f / 16.0f);
        ad *= (1.0f / 16.0f);
        for (int e = 0; e < NEL; ++e) {
            fb[e * F + D1     + c] = au; // tile_up
            fb[e * F + 2 * D1 + c] = ad; // tile_dn
        }
    }

    // ---------- phase 1b: ion map (8x64x32 GEMM, trivial -> scalar VALU) ----------
    {
        const int j = tid >> 5;          // 0..7
        const int c = tid & 31;          // 0..31
        const float* hi = h_ion + ((size_t)b * NION + j) * DION;
        float s = b_ion[c];
        #pragma unroll 8
        for (int k = 0; k < DION; ++k) s += hi[k] * W_ion[k * DEI + c];
        sHion[j * DEI + c] = fast_tanh(s);
    }

    // ---------- phase 1c: h_mapped via WMMA f32 16x16x4 (one (M,N) tile per wave) ----------
    {
        const int Mt = wave & 1;
        const int Nt = wave >> 1;
        const int row = Mt * 16 + l15;   // A: lanes 0-15/16-31 both map M = lane&15
        const int n   = Nt * 16 + l15;   // B/D: N = lane&15
        const float* arow = h1 + row * D1;
        v8f acc = {};
        #pragma unroll 8
        for (int kk = 0; kk < D1; kk += 4) {
            const int ka = kk + 2 * half;      // lanes 16-31 carry K=+2,+3
            v2f a;  a.x  = arow[ka];           a.y  = arow[ka + 1];
            v2f bb; bb.x = W_hmap[ka * EMB + n]; bb.y = W_hmap[(ka + 1) * EMB + n];
            acc = WMMA_F32X4(a, bb, acc);
        }
        const float bias = b_hmap[n];
        #pragma unroll
        for (int v = 0; v < 8; ++v) {          // D: VGPR v -> M = v + 8*half
            const int r = Mt * 16 + v + 8 * half;
            sHmap[r * EMB + n] = fast_tanh(acc[v] + bias);
        }
    }
    __syncthreads();

    // ---------- phase 2a: pair GEMM + tanh + * h_mapped[j] + store + j-sum ----------
    // 64 M-tiles (i, j-half) x 4 N-tiles; wave w owns M-tiles 8w..8w+7.
    for (int mtl = 0; mtl < 8; ++mtl) {
        const int  mt  = wave * 8 + mtl;
        const int  i   = mt >> 1;
        const int  j0  = (mt & 1) * 16;
        const bool same = ((i < 16) == (j0 == 0));   // uu/dd -> same, ud/du -> diff
        const float* Wp = same ? sWsame : sWdiff;
        const float* bp = same ? sBias[0] : sBias[1];

        // hoist A tile: rows are (i, j0 + lane&15), K = 0..31
        const float* arow = h_el_el + (((size_t)b * NEL + i) * NEL + (j0 + l15)) * D2;
        v2f a[8];
        #pragma unroll
        for (int s = 0; s < 8; ++s) {
            const int ka = 4 * s + 2 * half;
            a[s].x = arow[ka];
            a[s].y = arow[ka + 1];
        }

        float* orow = mult_out + (((size_t)b * NEL + i) * NEL) * EMB;

        for (int nt = 0; nt < 4; ++nt) {
            const int n = nt * 16 + l15;
            v8f acc = {};
            #pragma unroll
            for (int s = 0; s < 8; ++s) {
                const int ka = 4 * s + 2 * half;
                v2f bb; bb.x = Wp[ka * EMB + n]; bb.y = Wp[(ka + 1) * EMB + n];
                acc = WMMA_F32X4(a[s], bb, acc);
            }
            const float bias = bp[n];
            float partial = 0.0f;
            #pragma unroll
            for (int v = 0; v < 8; ++v) {
                const int j = j0 + v + 8 * half;
                const float m = fast_tanh(acc[v] + bias) * sHmap[j * EMB + n];
                orow[j * EMB + n] = m;         // el_el_mult output
                partial += m;
            }
            atomicAdd(&sAcc[i * EMB + n], partial);   // ds_add_f32, j-sum for el_el
        }
    }

    // ---------- phase 2b: f_pairs (means over j of h_el_el; L2-resident re-read) ----------
    {
        const int i  = tid >> 3;
        const int cg = (tid & 7) * 4;
        const float* pe = h_el_el + (((size_t)b * NEL + i) * NEL) * D2 + cg;
        float su[4] = {0, 0, 0, 0}, sd[4] = {0, 0, 0, 0};
        for (int j = 0; j < NEL; ++j) {
            const float4 v = *reinterpret_cast<const float4*>(pe + j * D2);
            if (j < 16) { su[0] += v.x; su[1] += v.y; su[2] += v.z; su[3] += v.w; }
            else        { sd[0] += v.x; sd[1] += v.y; sd[2] += v.z; sd[3] += v.w; }
        }
        #pragma unroll
        for (int q = 0; q < 4; ++q) {
            fb[i * F + 3 * D1      + cg + q] = su[q] * (1.0f / 16.0f);
            fb[i * F + 3 * D1 + 32 + cg + q] = sd[q] * (1.0f / 16.0f);
        }
    }

    // ---------- phase 2c: el_ion = mean_j h_el_ion * ion_map ----------
    {
        const int i  = tid >> 3;
        const int cg = (tid & 7) * 4;
        const float* pe = h_el_ion + (((size_t)b * NEL + i) * NION) * DEI + cg;
        float s[4] = {0, 0, 0, 0};
        #pragma unroll
        for (int j = 0; j < NION; ++j) {
            const float4 v = *reinterpret_cast<const float4*>(pe + j * DEI);
            s[0] += v.x * sHion[j * DEI + cg + 0];
            s[1] += v.y * sHion[j * DEI + cg + 1];
            s[2] += v.z * sHion[j * DEI + cg + 2];
            s[3] += v.w * sHion[j * DEI + cg + 3];
        }
        #pragma unroll
        for (int q = 0; q < 4; ++q)
            fb[i * F + 896 + cg + q] = s[q] * (1.0f / 8.0f);
    }
    __syncthreads();

    // ---------- phase 3: el_el = sAcc / 16 ----------
    {
        const int i  = tid >> 3;
        const int n0 = (tid & 7) * 8;
        #pragma unroll
        for (int q = 0; q < 8; ++q)
            fb[i * F + 832 + n0 + q] = sAcc[i * EMB + n0 + q] * (1.0f / 16.0f);
    }
}

extern "C" void kernel_launch(void* const* d_in, const int* in_sizes, int n_in,
                              void* d_out, int out_size, void* d_ws, size_t ws_size,
                              hipStream_t stream) {
    (void)in_sizes; (void)n_in; (void)out_size; (void)d_ws; (void)ws_size;

    const float* h_one    = (const float*)d_in[0];
    const float* h_ion    = (const float*)d_in[1];
    const float* h_el_el  = (const float*)d_in[2];
    const float* h_el_ion = (const float*)d_in[3];
    const float* W_hmap   = (const float*)d_in[4];
    const float* b_hmap   = (const float*)d_in[5];
    const float* W_wsame  = (const float*)d_in[6];
    const float* b_wsame  = (const float*)d_in[7];
    const float* W_wdiff  = (const float*)d_in[8];
    const float* b_wdiff  = (const float*)d_in[9];
    const float* W_ion    = (const float*)d_in[10];
    const float* b_ion    = (const float*)d_in[11];

    float* feats = (float*)d_out;
    float* mult  = feats + (size_t)NB * NEL * F;   // el_el_mult follows feats

    symfeat_kernel<<<dim3(NB), dim3(256), 0, stream>>>(
        h_one, h_ion, h_el_el, h_el_ion,
        W_hmap, b_hmap, W_wsame, b_wsame, W_wdiff, b_wdiff, W_ion, b_ion,
        feats, mult);
}